// AttnModel_4380866642114
// MI455X (gfx1250) — compile-verified
//
#include <hip/hip_runtime.h>
#include <hip/hip_bf16.h>

// Shapes (fixed by the reference)
#define Bsz 1024
#define Nn  256
#define Dd  256
#define Mm  768   // 3*D

typedef __attribute__((ext_vector_type(2))) float v2f;
typedef __attribute__((ext_vector_type(8))) float v8f;
typedef __attribute__((ext_vector_type(4))) unsigned int u32x4;
typedef __attribute__((ext_vector_type(4))) int          i32x4;
typedef __attribute__((ext_vector_type(8))) int          i32x8;

// ---------------------------------------------------------------------------
// Tensor Data Mover: DMA one 2-D tile (TILE_ROWS x 256 f32, row stride 256
// elements) from global memory into a contiguous LDS region.
// D# layout per CDNA5 ISA ch.8 (group0: count/lds/global/type, group1: sizes).
// ---------------------------------------------------------------------------
#define TILE_ROWS 8

__device__ __forceinline__ void tdm_load_tile(const float* gsrc, unsigned lds_off)
{
    const unsigned long long ga = (unsigned long long)gsrc;

    u32x4 g0;
    g0.x = 1u;                                            // count=1, is_restore=0
    g0.y = lds_off;                                       // LDS byte address
    g0.z = (unsigned)(ga & 0xffffffffull);                // global_addr[31:0]
    g0.w = (unsigned)((ga >> 32) & 0x1ffffffull)          // global_addr[56:32]
         | (2u << 30);                                    // type = 2 ("image")

    i32x8 g1;
    g1[0] = 0x00020000;                                   // data_size=2 (4 bytes), no multicast
    g1[1] = (int)(256u << 16);                            // tensor_dim0 = 256 (bits 79:48 lo)
    g1[2] = (int)((unsigned)TILE_ROWS << 16);             // tensor_dim1 = 8   (bits 95:80)
    g1[3] = (int)(256u << 16);                            // tile_dim0 = 256   (bits 127:112)
    g1[4] = TILE_ROWS;                                    // tile_dim1 = 8     (bits 143:128)
    g1[5] = 256;                                          // tensor_dim0_stride = 256 elements
    g1[6] = 0;
    g1[7] = 0;

    i32x4 gz = {0, 0, 0, 0};                              // groups 2/3 unused (2-D tile)
#if __clang_major__ >= 23
    i32x8 gz8 = {0, 0, 0, 0, 0, 0, 0, 0};
    __builtin_amdgcn_tensor_load_to_lds(g0, g1, gz, gz, gz8, 0);
#else
    __builtin_amdgcn_tensor_load_to_lds(g0, g1, gz, gz, 0);
#endif
}

// ---------------------------------------------------------------------------
// Kernel 1: fused attention (flash-style online softmax), one workgroup per
// batch row. K chunks (8 rows x 768) are DMA'd by the Tensor Data Mover into
// double-buffered LDS, overlapped with compute via TENSORcnt.
// Streams seq/seq_e/seq_t (805 MB total) exactly once.
// ---------------------------------------------------------------------------
__global__ __launch_bounds__(256) void attn_kernel(
    const float* __restrict__ src, const float* __restrict__ src_t,
    const float* __restrict__ seq, const float* __restrict__ seq_t,
    const float* __restrict__ seq_e, const unsigned char* __restrict__ mask,
    const float* __restrict__ shared_attn,
    float* __restrict__ y, float* __restrict__ attn_w)
{
    constexpr int CH  = TILE_ROWS;     // K rows per chunk
    constexpr int NCH = Nn / CH;       // 32 chunks

    __shared__ float wq_s[Mm];
    __shared__ float wk_s[Mm];
    __shared__ __attribute__((aligned(16))) float ks[2][CH * 256];  // seq   tiles
    __shared__ __attribute__((aligned(16))) float ke[2][CH * 256];  // seq_e tiles
    __shared__ __attribute__((aligned(16))) float kt[2][CH * 256];  // seq_t tiles
    __shared__ float s_all[Nn];        // masked logits (for attn_w output)
    __shared__ float e_ch[CH];
    __shared__ float red[8];
    __shared__ float sq_s;

    const int tid  = threadIdx.x;
    const int lane = tid & 31;
    const int wave = tid >> 5;         // 8 waves (wave32)
    const int b    = blockIdx.x;

    const size_t kbase = (size_t)b * Nn * Dd;   // element offset of this batch's K rows

    // Prefetch chunk 0 via TDM (wave 0 only; EXEC is ignored by tensor ops,
    // wave==0 is a wave-uniform branch so the descriptor stays scalar).
    if (wave == 0) {
        const size_t off = kbase;
        tdm_load_tile(seq   + off, (unsigned)(unsigned long long)&ks[0][0]);
        tdm_load_tile(seq_e + off, (unsigned)(unsigned long long)&ke[0][0]);
        tdm_load_tile(seq_t + off, (unsigned)(unsigned long long)&kt[0][0]);
    }

    // Stage wq/wk while the DMA flies.  wq = shared_attn[0:M], wk = [M:2M].
    for (int i = tid; i < 2 * Mm; i += 256) {
        float v = shared_attn[i];
        if (i < Mm) wq_s[i] = v; else wk_s[i - Mm] = v;
    }
    __syncthreads();

    // sq = src[b]·wq[0:256] + src_t[b]·wq[512:768]  (middle of q is zero)
    float p = src[(size_t)b * Dd + tid] * wq_s[tid]
            + src_t[(size_t)b * Dd + tid] * wq_s[512 + tid];
    for (int off = 16; off > 0; off >>= 1) p += __shfl_down(p, off, 32);
    if (lane == 0) red[wave] = p;
    __syncthreads();
    if (tid == 0) {
        float s = 0.f;
        for (int w = 0; w < 8; ++w) s += red[w];
        sq_s = s;
    }
    __syncthreads();
    const float sq = sq_s;

    // online softmax state
    float mx = -3.0e38f, Z = 0.f;
    float acc0 = 0.f, acc1 = 0.f, acc2 = 0.f;   // this thread's 3 output cols

    for (int c = 0; c < NCH; ++c) {
        const int d  = c & 1;
        const int n0 = c * CH;

        if (wave == 0) {
            if (c + 1 < NCH) {
                // issue next chunk into the other buffer, then wait for the
                // current chunk's 3 TDM ops (in-order per wave => cnt<=3).
                const size_t off = kbase + (size_t)(c + 1) * CH * Dd;
                tdm_load_tile(seq   + off, (unsigned)(unsigned long long)&ks[d ^ 1][0]);
                tdm_load_tile(seq_e + off, (unsigned)(unsigned long long)&ke[d ^ 1][0]);
                tdm_load_tile(seq_t + off, (unsigned)(unsigned long long)&kt[d ^ 1][0]);
                __builtin_amdgcn_s_wait_tensorcnt(3);
            } else {
                __builtin_amdgcn_s_wait_tensorcnt(0);
            }
        }
        __syncthreads();   // chunk c resident in buffer d for all waves

        // ---- logits: wave w reduces row w of the chunk ----
        {
            const float* pS = &ks[d][wave * 256];
            const float* pE = &ke[d][wave * 256];
            const float* pT = &kt[d][wave * 256];
            float dt = 0.f;
            for (int j = lane; j < 256; j += 32)
                dt += pS[j] * wk_s[j] + pE[j] * wk_s[256 + j] + pT[j] * wk_s[512 + j];
            for (int off = 16; off > 0; off >>= 1) dt += __shfl_down(dt, off, 32);
            if (lane == 0) {
                float s = sq + dt;
                if (mask[(size_t)b * Nn + n0 + wave]) s = -1.0e10f;
                s_all[n0 + wave] = s;
            }
        }
        __syncthreads();

        // ---- online-softmax update ----
        float cm = -3.0e38f;
        for (int r = 0; r < CH; ++r) cm = fmaxf(cm, s_all[n0 + r]);
        const float nm    = fmaxf(mx, cm);
        const float scale = __expf(mx - nm);
        if (tid < CH) e_ch[tid] = __expf(s_all[n0 + tid] - nm);
        __syncthreads();

        float cs = 0.f;
        for (int r = 0; r < CH; ++r) cs += e_ch[r];
        Z = Z * scale + cs;
        acc0 *= scale; acc1 *= scale; acc2 *= scale;
        for (int r = 0; r < CH; ++r) {
            const float e = e_ch[r];
            acc0 += e * ks[d][r * 256 + tid];
            acc1 += e * ke[d][r * 256 + tid];
            acc2 += e * kt[d][r * 256 + tid];
        }
        mx = nm;
        __syncthreads();   // done reading buffer d; wave0 may refill it next iter
    }

    const float inv = 1.0f / Z;
    y[(size_t)b * Mm +       tid] = acc0 * inv;
    y[(size_t)b * Mm + 256 + tid] = acc1 * inv;
    y[(size_t)b * Mm + 512 + tid] = acc2 * inv;
    attn_w[(size_t)b * Nn + tid] = __expf(s_all[tid] - mx) * inv;
}

// ---------------------------------------------------------------------------
// WMMA fp32 GEMM: C(rows x Nc) = A(rows x K) * W(Nc x K)^T  (both row-major).
// One wave computes a 32x32 tile (4 x 16x16 accumulators), stepping K by 4
// with V_WMMA_F32_16X16X4_F32.
//   EPI = 0: plain store   EPI = 1: add q=[src,0,src_t]   EPI = 2: ReLU
// ---------------------------------------------------------------------------
template <int EPI>
__global__ __launch_bounds__(128) void gemm_wmma_kernel(
    const float* __restrict__ A, const float* __restrict__ W,
    float* __restrict__ C, int K, int Nc,
    const float* __restrict__ q0, const float* __restrict__ q2)
{
    const int lane = threadIdx.x & 31;
    const int wave = threadIdx.x >> 5;
    const int tile = blockIdx.x * 4 + wave;
    const int tilesN = Nc >> 5;
    const int tm = tile / tilesN;
    const int tn = tile % tilesN;
    const int r0 = tm << 5;
    const int c0 = tn << 5;

    const int lrow = lane & 15;
    const int kb   = (lane >> 4) << 1;   // K sub-offset 0 or 2 (A/B frag layout)

    const float* a0 = A + (size_t)(r0 + lrow) * K + kb;
    const float* a1 = a0 + (size_t)16 * K;
    const float* b0 = W + (size_t)(c0 + lrow) * K + kb;
    const float* b1 = b0 + (size_t)16 * K;

    v8f acc00 = {}, acc01 = {}, acc10 = {}, acc11 = {};

#pragma unroll 4
    for (int k = 0; k < K; k += 4) {
        const v2f va0 = *(const v2f*)(a0 + k);
        const v2f va1 = *(const v2f*)(a1 + k);
        const v2f vb0 = *(const v2f*)(b0 + k);
        const v2f vb1 = *(const v2f*)(b1 + k);
        acc00 = __builtin_amdgcn_wmma_f32_16x16x4_f32(false, va0, false, vb0, (short)0, acc00, false, false);
        acc01 = __builtin_amdgcn_wmma_f32_16x16x4_f32(false, va0, false, vb1, (short)0, acc01, false, false);
        acc10 = __builtin_amdgcn_wmma_f32_16x16x4_f32(false, va1, false, vb0, (short)0, acc10, false, false);
        acc11 = __builtin_amdgcn_wmma_f32_16x16x4_f32(false, va1, false, vb1, (short)0, acc11, false, false);
    }

    // C/D layout: lane l, VGPR v -> (m = v + 8*(l/16), n = l%16)
    const int nloc  = lane & 15;
    const int mbase = (lane >> 4) << 3;

#pragma unroll
    for (int v = 0; v < 8; ++v) {
        const int m = mbase + v;
        float e00 = acc00[v], e01 = acc01[v], e10 = acc10[v], e11 = acc11[v];
        const int ra = r0 + m, rb = r0 + 16 + m;
        const int ca = c0 + nloc, cb = c0 + 16 + nloc;

        if (EPI == 1) {  // += q, q = [src, 0, src_t] along the 768-wide feature axis
            auto qv = [&](int row, int col) -> float {
                if (col < 256)      return q0[(size_t)row * Dd + col];
                else if (col < 512) return 0.f;
                else                return q2[(size_t)row * Dd + (col - 512)];
            };
            e00 += qv(ra, ca); e01 += qv(ra, cb);
            e10 += qv(rb, ca); e11 += qv(rb, cb);
        }
        if (EPI == 2) {
            e00 = fmaxf(e00, 0.f); e01 = fmaxf(e01, 0.f);
            e10 = fmaxf(e10, 0.f); e11 = fmaxf(e11, 0.f);
        }
        C[(size_t)ra * Nc + ca] = e00;
        C[(size_t)ra * Nc + cb] = e01;
        C[(size_t)rb * Nc + ca] = e10;
        C[(size_t)rb * Nc + cb] = e11;
    }
}

// ---------------------------------------------------------------------------
// Kernel 3: LayerNorm over M=768 of z (which already contains fc(out)+q),
// then write x = [LN(z), src]  (width 1024) for the aggregation GEMM.
// ---------------------------------------------------------------------------
__global__ __launch_bounds__(256) void ln_concat_kernel(
    const float* __restrict__ z, const float* __restrict__ src,
    const float* __restrict__ lnw, const float* __restrict__ lnb,
    float* __restrict__ x)
{
    __shared__ float red[8];
    __shared__ float mu_s, rs_s;
    const int tid = threadIdx.x, lane = tid & 31, wave = tid >> 5;
    const int b = blockIdx.x;

    const float* zr = z + (size_t)b * Mm;
    const float z0 = zr[tid], z1 = zr[tid + 256], z2 = zr[tid + 512];

    float s = z0 + z1 + z2;
    for (int off = 16; off > 0; off >>= 1) s += __shfl_down(s, off, 32);
    if (lane == 0) red[wave] = s;
    __syncthreads();
    if (tid == 0) {
        float t = 0.f; for (int w = 0; w < 8; ++w) t += red[w];
        mu_s = t * (1.0f / 768.0f);
    }
    __syncthreads();
    const float mu = mu_s;

    const float d0 = z0 - mu, d1 = z1 - mu, d2 = z2 - mu;
    float v = d0 * d0 + d1 * d1 + d2 * d2;
    for (int off = 16; off > 0; off >>= 1) v += __shfl_down(v, off, 32);
    if (lane == 0) red[wave] = v;
    __syncthreads();
    if (tid == 0) {
        float t = 0.f; for (int w = 0; w < 8; ++w) t += red[w];
        rs_s = rsqrtf(t * (1.0f / 768.0f) + 1e-5f);
    }
    __syncthreads();
    const float rs = rs_s;

    float* xr = x + (size_t)b * (Mm + Dd);
    xr[tid]       = d0 * rs * lnw[tid]       + lnb[tid];
    xr[tid + 256] = d1 * rs * lnw[tid + 256] + lnb[tid + 256];
    xr[tid + 512] = d2 * rs * lnw[tid + 512] + lnb[tid + 512];
    xr[768 + tid] = src[(size_t)b * Dd + tid];
}

// ---------------------------------------------------------------------------
extern "C" void kernel_launch(void* const* d_in, const int* in_sizes, int n_in,
                              void* d_out, int out_size, void* d_ws, size_t ws_size,
                              hipStream_t stream) {
    const float*         src     = (const float*)d_in[0];   // (B, D)
    const float*         src_t   = (const float*)d_in[1];   // (B, 1, D)
    const float*         seq     = (const float*)d_in[2];   // (B, N, D)
    const float*         seq_t   = (const float*)d_in[3];   // (B, N, D)
    const float*         seq_e   = (const float*)d_in[4];   // (B, N, D)
    const unsigned char* mask    = (const unsigned char*)d_in[5]; // (B, N) bool
    const float*         sh_attn = (const float*)d_in[6];   // (1, 2M)
    const float*         fc_w    = (const float*)d_in[7];   // (M, M)
    const float*         ln_w    = (const float*)d_in[8];   // (M,)
    const float*         ln_b    = (const float*)d_in[9];   // (M,)
    const float*         w1      = (const float*)d_in[10];  // (M, M+D)
    const float*         w2      = (const float*)d_in[11];  // (D, M)

    float* out    = (float*)d_out;            // (B, D)
    float* attn_w = out + (size_t)Bsz * Dd;   // (B, N)

    float* y = (float*)d_ws;                       // (B, M)
    float* z = y + (size_t)Bsz * Mm;               // (B, M)
    float* x = z + (size_t)Bsz * Mm;               // (B, M+D)
    float* u = x + (size_t)Bsz * (Mm + Dd);        // (B, M)

    // 1) fused attention (TDM-fed, double-buffered) -> y, attn_w
    attn_kernel<<<Bsz, 256, 0, stream>>>(src, src_t, seq, seq_t, seq_e, mask,
                                         sh_attn, y, attn_w);

    // 2) z = y @ fc_w^T + q     (1024x768 @ 768x768)
    gemm_wmma_kernel<1><<<(32 * 24) / 4, 128, 0, stream>>>(y, fc_w, z, Mm, Mm, src, src_t);

    // 3) x = [LN(z), src]
    ln_concat_kernel<<<Bsz, 256, 0, stream>>>(z, src, ln_w, ln_b, x);

    // 4) u = relu(x @ w1^T)     (1024x1024 @ 1024x768)
    gemm_wmma_kernel<2><<<(32 * 24) / 4, 128, 0, stream>>>(x, w1, u, Mm + Dd, Mm, nullptr, nullptr);

    // 5) out = u @ w2^T         (1024x768 @ 768x256)
    gemm_wmma_kernel<0><<<(32 * 8) / 4, 128, 0, stream>>>(u, w2, out, Mm, Dd, nullptr, nullptr);
}